// ObstacleToLaneRelation_25675314495800
// MI455X (gfx1250) — compile-verified
//
#include <hip/hip_runtime.h>
#include <hip/hip_bf16.h>
#include <stdint.h>

// Problem geometry (from the reference): lane_features (M,150,4) f32,
// obs_pos (N,2) f32, same_obs_mask (M,1) i32.
#define NODE 150
#define FEATS 4
#define WAVES_PER_BLOCK 8

typedef float fvec2 __attribute__((ext_vector_type(2)));

// ---- CDNA5 async global->LDS DMA (gfx1250) ------------------------------
// Per-lane: copies 16 bytes from the lane's global address to its LDS address. Tracked by ASYNCcnt.
// lane_features is a 480MB one-shot stream (2.5x the 192MB L2) -> mark the
// traffic non-temporal so it does not evict the reusable obs_pos gather data.
__device__ __forceinline__ void async_ld_b128_nt(const float* gptr, float* lds_generic) {
    unsigned lds_addr = (unsigned)(uintptr_t)lds_generic;      // WG-relative LDS byte addr
    unsigned long long gaddr = (unsigned long long)(uintptr_t)gptr;
    asm volatile("global_load_async_to_lds_b128 %0, %1, off th:TH_LOAD_NT"
                 :: "v"(lds_addr), "v"(gaddr)
                 : "memory");
}

__device__ __forceinline__ void wait_asynccnt0() {
    asm volatile("s_wait_asynccnt 0" ::: "memory");
}

__global__ __launch_bounds__(256) void obstlane_kernel(
    const float* __restrict__ lf,     // (M,150,4)
    const float* __restrict__ obs,    // (N,2)
    const int*   __restrict__ mask,   // (M,1)
    float* __restrict__ proj,         // (M,2)
    float* __restrict__ idxo,         // (M,2) as float
    float* __restrict__ repo,         // (M,2)
    int M)
{
    __shared__ float smem[WAVES_PER_BLOCK * NODE * FEATS];  // 19200 B / block

    const int wave = threadIdx.x >> 5;
    const int lane = threadIdx.x & 31;
    const long long row = (long long)blockIdx.x * WAVES_PER_BLOCK + wave;
    if (row >= M) return;   // per-wave uniform

    float* lrow = smem + wave * (NODE * FEATS);
    const float* grow = lf + row * (long long)(NODE * FEATS);

    // --- Stage the whole row (150 x 16B) into LDS via async DMA ---
    // 5 issues: 32,32,32,32,22 active lanes; each active lane moves 16B.
    #pragma unroll
    for (int k = lane; k < NODE; k += 32)
        async_ld_b128_nt(grow + k * FEATS, lrow + k * FEATS);

    // Overlap the DMA with the (independent, cache-friendly) rep gather.
    int oi = mask[row];
    float rx = obs[2 * oi + 0];
    float ry = obs[2 * oi + 1];

    wait_asynccnt0();

    // --- Argmin over nodes 1..148 of squared distance to rep ---
    float bd = 3.402823e38f;
    int   bi = 1;
    #pragma unroll
    for (int k = 1 + lane; k <= NODE - 2; k += 32) {
        float x = lrow[k * FEATS + 0];
        float y = lrow[k * FEATS + 1];
        float dx = x - rx, dy = y - ry;
        float d = dx * dx + dy * dy;
        if (d < bd) { bd = d; bi = k; }   // ascending k keeps lowest index per lane
    }
    // wave32 reduction; tie-break on lower index (matches jnp.argmin)
    #pragma unroll
    for (int off = 16; off > 0; off >>= 1) {
        float od = __shfl_xor(bd, off, 32);
        int   ob = __shfl_xor(bi, off, 32);
        if (od < bd || (od == bd && ob < bi)) { bd = od; bi = ob; }
    }
    const int mi = bi;   // in [1,148]

    // --- Neighbor comparison over all 4 features (LDS broadcast reads) ---
    const float4* nd = (const float4*)lrow;
    float4 c = nd[mi];
    float4 p = nd[mi - 1];
    float4 n = nd[mi + 1];
    float dpv = (p.x - c.x) * (p.x - c.x) + (p.y - c.y) * (p.y - c.y)
              + (p.z - c.z) * (p.z - c.z) + (p.w - c.w) * (p.w - c.w);
    float dnv = (n.x - c.x) * (n.x - c.x) + (n.y - c.y) * (n.y - c.y)
              + (n.z - c.z) * (n.z - c.z) + (n.w - c.w) * (n.w - c.w);

    int ib, ia;
    float sx, sy, ex, ey;
    if (dnv < dpv) { ib = mi;     ia = mi + 1; sx = c.x; sy = c.y; ex = n.x; ey = n.y; }
    else           { ib = mi - 1; ia = mi;     sx = p.x; sy = p.y; ex = c.x; ey = c.y; }

    // --- Projection of rep onto segment [s,e] ---
    float vx = ex - sx, vy = ey - sy;
    float mg = sqrtf(vx * vx + vy * vy);
    float ux = vx / mg, uy = vy / mg;
    float t  = (rx - sx) * ux + (ry - sy) * uy;
    float px = sx + t * ux;
    float py = sy + t * uy;

    if (lane == 0) {
        // Write-once output streams: non-temporal stores.
        fvec2 vproj = { px, py };
        fvec2 vidx  = { (float)ib, (float)ia };
        fvec2 vrep  = { rx, ry };
        __builtin_nontemporal_store(vproj, (fvec2*)(proj + 2 * row));
        __builtin_nontemporal_store(vidx,  (fvec2*)(idxo + 2 * row));
        __builtin_nontemporal_store(vrep,  (fvec2*)(repo + 2 * row));
    }
}

extern "C" void kernel_launch(void* const* d_in, const int* in_sizes, int n_in,
                              void* d_out, int out_size, void* d_ws, size_t ws_size,
                              hipStream_t stream) {
    const float* lf   = (const float*)d_in[0];   // (M,150,4)
    const float* obs  = (const float*)d_in[1];   // (N,2)
    const int*   mask = (const int*)d_in[2];     // (M,1)
    const int M = in_sizes[2];                   // flat count of mask == M

    float* out  = (float*)d_out;
    float* proj = out;                           // M*2
    float* idxo = out + (size_t)2 * M;           // M*2 (ints as float)
    float* repo = out + (size_t)4 * M;           // M*2

    const int blocks = (M + WAVES_PER_BLOCK - 1) / WAVES_PER_BLOCK;
    hipLaunchKernelGGL(obstlane_kernel, dim3(blocks), dim3(256), 0, stream,
                       lf, obs, mask, proj, idxo, repo, M);
}